// MultiheadAttention_49984829390901
// MI455X (gfx1250) — compile-verified
//
#include <hip/hip_runtime.h>
#include <math.h>

// ---- CDNA5 WMMA vector types (fp32 16x16x4: A=v2f, B=v2f, C/D=v8f) ----
typedef __attribute__((ext_vector_type(2))) float v2f;
typedef __attribute__((ext_vector_type(4))) float v4f;
typedef __attribute__((ext_vector_type(8))) float v8f;

// Problem constants (match reference)
constexpr int   BATCH     = 32;
constexpr int   T         = 1024;
constexpr int   DK        = 64;
constexpr float SCALE     = 0.125f;            // 1/sqrt(64)
constexpr float MASK_FILL = -4294967295.0f;    // -2^32 + 1

// LDS padded strides. Both are 16-byte-aligned multiples (1028*4, 68*4)
// and stride % 64 == 4 -> conflict-free per-row lane access for WMMA A reads.
constexpr int SP = T + 4;   // score rows: 16 x 1028 floats
constexpr int QP = DK + 4;  // Q rows:     16 x 68 floats

__global__ __launch_bounds__(128)
void attn_fwd_kernel(const float* __restrict__ Kmat,
                     const float* __restrict__ Vmat,
                     const float* __restrict__ Qmat,
                     const unsigned char* __restrict__ mask,
                     const float* __restrict__ qmask,
                     float* __restrict__ outO,   // [B,T,DK]
                     float* __restrict__ outP)   // [B,T,T]
{
    __shared__ float sS[16 * SP];     // 65,792 B  score / prob tile (16 rows x T)
    __shared__ float sQ[16 * QP];     //  4,352 B  Q tile
    __shared__ float red[128];        // per-thread reduction scratch
    __shared__ float rowScale[16];    // row max, then (query_mask / sum)

    const int b    = blockIdx.y;
    const int q0   = blockIdx.x * 16;
    const int tid  = threadIdx.x;
    const int wave = tid >> 5;        // 0..3
    const int lane = tid & 31;
    const int n    = lane & 15;       // N column / A row within half
    const int half = lane >> 4;       // 0 or 1
    const int dd   = 2 * half;        // K-chunk offset owned by this half

    const float* Qb = Qmat + ((size_t)b * T + q0) * DK;
    const float* Kb = Kmat + (size_t)b * T * DK;
    const float* Vb = Vmat + (size_t)b * T * DK;

    // ---------- Phase 0: stage Q tile into LDS (b128 loads/stores) ----------
    {
        const v4f* Qv = (const v4f*)Qb;                 // 256 float4s
        #pragma unroll
        for (int i = 0; i < (16 * DK) / (128 * 4); ++i) {
            const int  idx = i * 128 + tid;             // vec4 index
            const int  r   = (idx * 4) >> 6;
            const int  c   = (idx * 4) & 63;
            *(v4f*)&sQ[r * QP + c] = Qv[idx];
        }
    }
    __syncthreads();

    // ---------- Phase 1: S = (Q K^T) * scale, masked -> LDS ----------
    // Each wave computes every 4th 16-wide key tile.
    for (int kt = wave; kt < T / 16; kt += 4) {
        const int k0 = kt * 16;
        v8f c = {};
        const float* Krow = Kb + (size_t)(k0 + n) * DK;  // B operand: K[k0+n][*]
        #pragma unroll
        for (int dc = 0; dc < DK / 4; ++dc) {
            const int d0 = dc * 4;
            v2f a, bv;
            a.x  = sQ[n * QP + d0 + dd];
            a.y  = sQ[n * QP + d0 + dd + 1];
            bv.x = Krow[d0 + dd];
            bv.y = Krow[d0 + dd + 1];
            c = __builtin_amdgcn_wmma_f32_16x16x4_f32(
                    false, a, false, bv, (short)0, c, false, false);
        }
        const size_t mbase = ((size_t)b * T + q0) * T;  // mask[b][q][k]
        #pragma unroll
        for (int r = 0; r < 8; ++r) {
            const int m   = r + 8 * half;
            const int col = k0 + n;
            float s = c[r] * SCALE;
            // mask bytes are single-use: NT load keeps them out of the way in L2
            unsigned char mb =
                __builtin_nontemporal_load(&mask[mbase + (size_t)m * T + col]);
            if (mb) s = MASK_FILL;
            sS[m * SP + col] = s;
        }
    }
    __syncthreads();

    // ---------- Phase 2: softmax over each of 16 rows (8 threads/row) ----------
    {
        const int r  = tid >> 3;   // row 0..15
        const int t8 = tid & 7;

        float mx = -INFINITY;
        for (int j = t8; j < T; j += 8) mx = fmaxf(mx, sS[r * SP + j]);
        red[tid] = mx;
        __syncthreads();
        if (t8 == 0) {
            float m2 = red[tid];
            #pragma unroll
            for (int i = 1; i < 8; ++i) m2 = fmaxf(m2, red[tid + i]);
            rowScale[r] = m2;
        }
        __syncthreads();
        const float M = rowScale[r];

        float sum = 0.f;
        for (int j = t8; j < T; j += 8) {
            float e = __expf(sS[r * SP + j] - M);
            sS[r * SP + j] = e;
            sum += e;
        }
        red[tid] = sum;
        __syncthreads();
        if (t8 == 0) {
            float s2 = 0.f;
            #pragma unroll
            for (int i = 0; i < 8; ++i) s2 += red[tid + i];
            // fold post-softmax query mask + normalization into one scale
            rowScale[r] = qmask[(size_t)b * T + q0 + r] / s2;
        }
        __syncthreads();
    }

    // ---------- Phase 3: scale probs in LDS, stream attn to HBM ----------
    // b128 nontemporal stores: this 128 MB write-once stream must not evict
    // the K/V working set (16 MB, fully resident in the 192 MB L2).
    for (int r2 = 0; r2 < 16; ++r2) {
        const float f = rowScale[r2];
        float* __restrict__ Prow = outP + ((size_t)b * T + q0 + r2) * T;
        #pragma unroll
        for (int jj = 0; jj < T / (128 * 4); ++jj) {    // 2 iterations
            const int j = (jj * 128 + tid) * 4;
            v4f v = *(v4f*)&sS[r2 * SP + j];
            v.x *= f; v.y *= f; v.z *= f; v.w *= f;
            *(v4f*)&sS[r2 * SP + j] = v;                // keep for P@V
            __builtin_nontemporal_store(v, (v4f*)(Prow + j));
        }
    }
    __syncthreads();

    // ---------- Phase 4: O = P @ V (each wave owns a 16-wide d slice) ----------
    {
        const int d0w = wave * 16;
        v8f c = {};
        for (int kk = 0; kk < T / 4; ++kk) {
            const int k0 = kk * 4;
            v2f a, bv;
            a.x = sS[n * SP + k0 + dd];
            a.y = sS[n * SP + k0 + dd + 1];
            const float* Vp = Vb + (size_t)(k0 + dd) * DK + d0w + n;
            bv.x = Vp[0];
            bv.y = Vp[DK];
            c = __builtin_amdgcn_wmma_f32_16x16x4_f32(
                    false, a, false, bv, (short)0, c, false, false);
        }
        float* __restrict__ Ob = outO + ((size_t)b * T + q0) * DK;
        #pragma unroll
        for (int r = 0; r < 8; ++r) {
            const int m = r + 8 * half;
            Ob[(size_t)m * DK + d0w + n] = c[r];
        }
    }
}

extern "C" void kernel_launch(void* const* d_in, const int* in_sizes, int n_in,
                              void* d_out, int out_size, void* d_ws, size_t ws_size,
                              hipStream_t stream) {
    // setup_inputs order: key, value, query, mask(bool), query_mask
    const float*         Kmat  = (const float*)d_in[0];
    const float*         Vmat  = (const float*)d_in[1];
    const float*         Qmat  = (const float*)d_in[2];
    const unsigned char* mask  = (const unsigned char*)d_in[3];  // numpy bool = 1 byte
    const float*         qmask = (const float*)d_in[4];

    float* outO = (float*)d_out;                               // result [B,T,DK]
    float* outP = (float*)d_out + (size_t)BATCH * T * DK;      // attn   [B,T,T]

    dim3 grid(T / 16, BATCH);   // 64 x 32 workgroups
    dim3 block(128);            // 4 waves (wave32)
    attn_fwd_kernel<<<grid, block, 0, stream>>>(Kmat, Vmat, Qmat, mask, qmask, outO, outP);
}